// GraphConv_30889404793461
// MI455X (gfx1250) — compile-verified
//
#include <hip/hip_runtime.h>
#include <hip/hip_bf16.h>
#include <math.h>

typedef float v2f __attribute__((ext_vector_type(2)));
typedef float v8f __attribute__((ext_vector_type(8)));

#define N_FEAT  128
#define N_CLASS 64

// ---------------------------------------------------------------- zero fill
__global__ void gcn_zero_kernel(float* __restrict__ p, long long n) {
    long long i = (long long)blockIdx.x * blockDim.x + threadIdx.x;
    long long stride = (long long)gridDim.x * blockDim.x;
    for (; i < n; i += stride) p[i] = 0.0f;
}

// ---------------------------------------------------------------- weighted in-degree
__global__ void gcn_deg_kernel(const int* __restrict__ col,
                               const float* __restrict__ w,
                               float* __restrict__ deg, int E) {
    int e = blockIdx.x * blockDim.x + threadIdx.x;
    if (e < E) atomicAdd(&deg[col[e]], w[e]);
}

// deg -> dis = rsqrt(deg + 1)   (self-loop weight 1 included; always > 0)
__global__ void gcn_dis_kernel(float* __restrict__ deg, int n) {
    int i = blockIdx.x * blockDim.x + threadIdx.x;
    if (i < n) deg[i] = rsqrtf(deg[i] + 1.0f);
}

// ---------------------------------------------------------------- h = x @ W via V_WMMA_F32_16X16X4_F32
// One wave32 computes a 16x16 tile of h. blockDim = 256 (8 waves).
// Tile grid: (ceil(n/16)) x (N_CLASS/16). K loop: N_FEAT/4 = 32 WMMAs.
__global__ __launch_bounds__(256) void gcn_gemm_wmma_kernel(
        const float* __restrict__ x,   // [n, 128]
        const float* __restrict__ W,   // [128, 64]
        float* __restrict__ h,         // [n, 64]
        int n, int tilesM) {
    const int lane = threadIdx.x & 31;
    const int wave = threadIdx.x >> 5;
    const long long tile = (long long)blockIdx.x * 8 + wave;
    const int tilesN = N_CLASS / 16;                 // 4
    if (tile >= (long long)tilesM * tilesN) return;  // wave-uniform guard
    const int tileN = (int)(tile % tilesN);
    const int tileM = (int)(tile / tilesN);
    const int m0 = tileM * 16;
    const int n0 = tileN * 16;

    // A layout (16x4 f32): lanes 0-15 hold K={k0,k0+1}, lanes 16-31 K={k0+2,k0+3}, M = lane&15
    // B layout (4x16 f32): VGPR r, lanes 0-15 -> K=k0+r, lanes 16-31 -> K=k0+r+2, N = lane&15
    const int khalf = (lane >> 4) * 2;                       // 0 or 2
    int mrow = m0 + (lane & 15);
    if (mrow >= n) mrow = n - 1;                             // clamp load, keep EXEC full
    const int ncol = n0 + (lane & 15);

    v8f c = {};
    const float* __restrict__ arow = x + (long long)mrow * N_FEAT + khalf;
    #pragma unroll
    for (int k0 = 0; k0 < N_FEAT; k0 += 4) {
        v2f a, b;
        a.x = arow[k0 + 0];
        a.y = arow[k0 + 1];
        const float* bp = W + (long long)(k0 + khalf) * N_CLASS + ncol;
        b.x = bp[0];
        b.y = bp[N_CLASS];
        // D = A*B + C  (8 args: neg_a, A, neg_b, B, c_mod, C, reuse_a, reuse_b)
        c = __builtin_amdgcn_wmma_f32_16x16x4_f32(false, a, false, b,
                                                  (short)0, c, false, false);
    }

    // D layout: VGPR r, lanes 0-15 -> M=m0+r, lanes 16-31 -> M=m0+r+8, N = n0+(lane&15)
    const int mbase = m0 + (lane >> 4) * 8;
    #pragma unroll
    for (int r = 0; r < 8; ++r) {
        const int m = mbase + r;
        if (m < n) h[(long long)m * N_CLASS + ncol] = c[r];
    }
}

// ---------------------------------------------------------------- edge scatter (one wave32 per edge)
// lane j handles features j and j+32; h rows are 256B -> fully coalesced;
// atomics land in L2 (h + out working sets ~51MB << 192MB L2 on MI455X).
__global__ __launch_bounds__(256) void gcn_scatter_kernel(
        const int* __restrict__ row, const int* __restrict__ col,
        const float* __restrict__ w, const float* __restrict__ dis,
        const float* __restrict__ h, float* __restrict__ out, int E) {
    const int lane = threadIdx.x & 31;
    const long long e = ((long long)blockIdx.x * blockDim.x + threadIdx.x) >> 5;
    if (e >= E) return;
    const int r = row[e];
    const int cdst = col[e];
    const float norm = dis[r] * w[e] * dis[cdst];
    const float* __restrict__ hr = h + (long long)r * N_CLASS;
    float* __restrict__ oc = out + (long long)cdst * N_CLASS;
    atomicAdd(&oc[lane],      hr[lane]      * norm);
    atomicAdd(&oc[lane + 32], hr[lane + 32] * norm);
}

// ---------------------------------------------------------------- self-loop + bias + sigmoid
__global__ void gcn_finalize_kernel(const float* __restrict__ h,
                                    const float* __restrict__ dis,
                                    const float* __restrict__ b,
                                    float* __restrict__ out, long long total) {
    const long long idx = (long long)blockIdx.x * blockDim.x + threadIdx.x;
    if (idx >= total) return;
    const int i = (int)(idx >> 6);   // node
    const int j = (int)(idx & 63);   // class
    const float d = dis[i];
    const float v = out[idx] + h[idx] * d * d + b[j];
    out[idx] = 1.0f / (1.0f + expf(-v));
}

// ---------------------------------------------------------------- launcher
extern "C" void kernel_launch(void* const* d_in, const int* in_sizes, int n_in,
                              void* d_out, int out_size, void* d_ws, size_t ws_size,
                              hipStream_t stream) {
    const float* x          = (const float*)d_in[0];   // [n, 128]
    const int*   edge_index = (const int*)  d_in[1];   // [2, E] flat
    const float* ew         = (const float*)d_in[2];   // [E]
    const float* W          = (const float*)d_in[3];   // [128, 64]
    const float* b          = (const float*)d_in[4];   // [64]

    const int n = in_sizes[0] / N_FEAT;
    const int E = in_sizes[2];
    const int* row = edge_index;
    const int* col = edge_index + E;
    float* out = (float*)d_out;

    // workspace layout: dis[n] | h[n * 64]
    float* dis = (float*)d_ws;
    float* h   = dis + n;

    const long long outTotal = (long long)n * N_CLASS;

    // 1) zero degree accumulator and output accumulator
    gcn_zero_kernel<<<1024, 256, 0, stream>>>(dis, n);
    gcn_zero_kernel<<<4096, 256, 0, stream>>>(out, outTotal);

    // 2) weighted in-degree over edges
    gcn_deg_kernel<<<(E + 255) / 256, 256, 0, stream>>>(col, ew, dis, E);

    // 3) dis = rsqrt(deg + 1)
    gcn_dis_kernel<<<(n + 255) / 256, 256, 0, stream>>>(dis, n);

    // 4) h = x @ W  (WMMA f32 16x16x4)
    const int tilesM = (n + 15) / 16;
    const long long waves = (long long)tilesM * (N_CLASS / 16);
    const unsigned gemmBlocks = (unsigned)((waves + 7) / 8);
    gcn_gemm_wmma_kernel<<<gemmBlocks, 256, 0, stream>>>(x, W, h, n, tilesM);

    // 5) edge scatter: out[col] += h[row] * norm
    const long long scatterThreads = (long long)E * 32;
    const unsigned scatterBlocks = (unsigned)((scatterThreads + 255) / 256);
    gcn_scatter_kernel<<<scatterBlocks, 256, 0, stream>>>(row, col, ew, dis, h, out, E);

    // 6) out = sigmoid(out + h * dis^2 + b)
    const unsigned finBlocks = (unsigned)((outTotal + 255) / 256);
    gcn_finalize_kernel<<<finBlocks, 256, 0, stream>>>(h, dis, b, out, outTotal);
}